// HierarchyModel_46445776339038
// MI455X (gfx1250) — compile-verified
//
#include <hip/hip_runtime.h>
#include <math.h>

#define NN 2048
#define DD 128
#define CC 8192
#define SS 16
#define HH 32
#define EPSF 1e-10f
#define CIRCLEF 2.0f
#define CLIP_MAXF 10000000000.0f

typedef __attribute__((ext_vector_type(2))) float v2f;
typedef __attribute__((ext_vector_type(8))) float v8f;

// ws float layout:
//   [0      .. 2047]  inner[i] = sum_d omega[i,d]^2
//   [2048   .. 4095]  colsum_od[j]   (column sums of clipped omegaDist)
//   [4096   .. 6143]  colsum_rd[j]   (column sums of realD)
//   [6144]  ss_od   [6145] ss_rd   [6146] overlap
//   [6147]  exceed  [6148] shape   [6149] positive
#define WS_INNER 0
#define WS_CSOD  2048
#define WS_CSRD  4096
#define WS_SCAL  6144

__device__ __forceinline__ float blockReduce256(float v, float* red) {
    int tid = threadIdx.x;
    red[tid] = v;
    __syncthreads();
    for (int s = 128; s > 0; s >>= 1) {
        if (tid < s) red[tid] += red[tid + s];
        __syncthreads();
    }
    float r = red[0];
    __syncthreads();
    return r;
}

// ---------------- inner[i] = ||omega_i||^2 ----------------
__global__ __launch_bounds__(256) void inner_kernel(const float* __restrict__ omega,
                                                    float* __restrict__ inner) {
    int i = blockIdx.x * blockDim.x + threadIdx.x;
    if (i < NN) {
        const float4* p = (const float4*)(omega + (size_t)i * DD);
        float s = 0.f;
        #pragma unroll 8
        for (int d = 0; d < DD / 4; ++d) {
            float4 q = p[d];
            s += q.x * q.x + q.y * q.y + q.z * q.z + q.w * q.w;
        }
        inner[i] = s;
    }
}

// ---------------- WMMA GEMM: omegaDist stats ----------------
// Each wave computes a 16(M) x 64(N) strip of G = Omega * Omega^T using
// v_wmma_f32_16x16x4_f32 (K-loop of 32), then folds od = max(-2G+inn_i+inn_j,EPS)
// into ss_od (Frobenius^2) and colsum_od (per-column sums) without storing G.
__global__ __launch_bounds__(256) void omega_gemm_kernel(const float* __restrict__ omega,
                                                         const float* __restrict__ inner,
                                                         float* __restrict__ colsum_od,
                                                         float* __restrict__ scal) {
    const int lane = threadIdx.x & 31;
    const int wave = blockIdx.x * 8 + (threadIdx.x >> 5);   // 4096 waves total
    const int mi = wave >> 5;          // 0..127 : 16-row tile
    const int nj = wave & 31;          // 0..31  : 64-col strip
    const int i0 = mi * 16;
    const int j0 = nj * 64;
    const int half = lane >> 4;        // K-pair select (A layout: lanes 16-31 hold K=2,3)
    const int lp   = lane & 15;

    const float* a_base = omega + (size_t)(i0 + lp) * DD + 2 * half;
    const float* b_base = omega + (size_t)(j0 + lp) * DD + 2 * half;

    v8f acc[4] = {};
    for (int k = 0; k < DD; k += 4) {
        v2f a = *(const v2f*)(a_base + k);
        #pragma unroll
        for (int t = 0; t < 4; ++t) {
            v2f b = *(const v2f*)(b_base + (size_t)t * (16 * DD) + k);
            // D = A(16x4, f32) * B(4x16, f32) + C ; full f32 accumulate
            acc[t] = __builtin_amdgcn_wmma_f32_16x16x4_f32(
                false, a, false, b, (short)0, acc[t], false, false);
        }
    }

    // rows covered by this lane: i0 + v + 8*half, v = 0..7 (C/D layout)
    float inn_i[8];
    #pragma unroll
    for (int v = 0; v < 8; ++v) inn_i[v] = inner[i0 + v + 8 * half];

    float ss = 0.f;
    #pragma unroll
    for (int t = 0; t < 4; ++t) {
        int gj = j0 + t * 16 + lp;     // column owned by this lane (N = lane%16)
        float innj = inner[gj];
        float colp = 0.f;
        #pragma unroll
        for (int v = 0; v < 8; ++v) {
            float od = fmaxf(-2.0f * acc[t][v] + inn_i[v] + innj, EPSF);
            ss += od * od;
            colp += od;
        }
        atomicAdd(&colsum_od[gj], colp);
    }
    // wave32 reduction for ss, one atomic per wave
    for (int off = 16; off > 0; off >>= 1) ss += __shfl_xor(ss, off, 32);
    if (lane == 0) atomicAdd(&scal[0], ss);
}

// ---------------- pairwise: overlap + realD stats ----------------
__global__ __launch_bounds__(256) void pairwise_kernel(const int* __restrict__ idIdx,
                                                       const float* __restrict__ childEmb,
                                                       const float* __restrict__ layerDist,
                                                       float* __restrict__ colsum_rd,
                                                       float* __restrict__ scal) {
    __shared__ float rlow[16][17], rhigh[16][17], rinv[16][17];
    __shared__ float clow[16][17], chigh[16][17];
    __shared__ int   rid[16], cid[16];
    __shared__ float colpart[16];
    __shared__ float red[256];

    const int tx = threadIdx.x & 15;   // local j
    const int ty = threadIdx.x >> 4;   // local i
    const int i0 = blockIdx.y * 16;
    const int j0 = blockIdx.x * 16;

    if (threadIdx.x < 16) {
        rid[threadIdx.x] = idIdx[i0 + threadIdx.x];
        cid[threadIdx.x] = idIdx[j0 + threadIdx.x];
        colpart[threadIdx.x] = 0.f;
    }
    __syncthreads();
    {   // stage 16x16 low/high for row block and column block (1 elem/thread each)
        float lo = childEmb[(size_t)rid[ty] * HH + tx];
        float hi = childEmb[(size_t)rid[ty] * HH + SS + tx];
        rlow[ty][tx] = lo; rhigh[ty][tx] = hi;
        rinv[ty][tx] = 1.0f / fmaxf(hi - lo, EPSF);
        clow[ty][tx]  = childEmb[(size_t)cid[ty] * HH + tx];
        chigh[ty][tx] = childEmb[(size_t)cid[ty] * HH + SS + tx];
    }
    __syncthreads();

    float l1 = 0.f, ov = 0.f;
    #pragma unroll
    for (int s = 0; s < SS; ++s) {
        float li = rlow[ty][s], hi = rhigh[ty][s];
        float lj = clow[tx][s], hj = chigh[tx][s];
        l1 += fabsf(li - lj);
        float o = fminf(hi, hj) - fmaxf(li, lj);
        ov += fmaxf(o, 0.f) * rinv[ty][s];
    }
    if (i0 + ty == j0 + tx) ov = 0.f;  // mask = 1 - eye(N)

    float acc   = layerDist[(size_t)rid[ty] * CC + cid[tx]];
    float realD = acc + l1;

    atomicAdd(&colpart[tx], realD);                 // v[j] = sum_i realD[i,j]
    float ssb = blockReduce256(realD * realD, red); // ||realD||_F^2 partial
    float ovb = blockReduce256(ov, red);
    if (threadIdx.x == 0) {
        atomicAdd(&scal[1], ssb);
        atomicAdd(&scal[2], ovb);
    }
    __syncthreads();
    if (threadIdx.x < 16) atomicAdd(&colsum_rd[j0 + threadIdx.x], colpart[threadIdx.x]);
}

// ---------------- elementwise N x 16 losses ----------------
__global__ __launch_bounds__(256) void elemwise_kernel(const int* __restrict__ idIdx,
                                                       const int* __restrict__ parentIds,
                                                       const float* __restrict__ childEmb,
                                                       const float* __restrict__ res,
                                                       const float* __restrict__ leaves,
                                                       float* __restrict__ scal) {
    __shared__ float red[256];
    int gid = blockIdx.x * 256 + threadIdx.x;      // < N*S = 32768
    int i = gid >> 4, s = gid & 15;
    int ci = idIdx[i], pi = parentIds[i];
    float low  = childEmb[(size_t)ci * HH + s];
    float high = childEmb[(size_t)ci * HH + SS + s];
    float rl = res[(size_t)pi * HH + s];
    float rh = res[(size_t)pi * HH + SS + s];

    float exceed = fmaxf(rl + CIRCLEF - low, 0.f) + fmaxf(high - (rh + CIRCLEF), 0.f);
    float diff = high - low;
    float numer = fmaxf(diff / (rh - rl), EPSF);
    float shape = fminf(fabsf(logf(numer / leaves[ci])), CLIP_MAXF);
    float pos = fmaxf(expf(-diff), EPSF);

    float e = blockReduce256(exceed, red);
    float sh = blockReduce256(shape, red);
    float p = blockReduce256(pos, red);
    if (threadIdx.x == 0) {
        atomicAdd(&scal[3], e);
        atomicAdd(&scal[4], sh);
        atomicAdd(&scal[5], p);
    }
}

// ---------------- combine ----------------
// lossDist^2 = SSod/a^2 - 2*dot/(a*b) + N*sumv2/b^2
//   a = max(||omegaDist||,EPS), b = max(||realD||,EPS),
//   dot = sum_j colsum_od[j]*v[j], sumv2 = sum_j v[j]^2
__global__ __launch_bounds__(256) void final_kernel(const float* __restrict__ colsum_od,
                                                    const float* __restrict__ colsum_rd,
                                                    const float* __restrict__ scal,
                                                    float* __restrict__ out) {
    __shared__ float red[256];
    float dot = 0.f, sv2 = 0.f;
    for (int j = threadIdx.x; j < NN; j += 256) {
        float v = colsum_rd[j];
        dot += colsum_od[j] * v;
        sv2 += v * v;
    }
    float dotr = blockReduce256(dot, red);
    float sv2r = blockReduce256(sv2, red);
    if (threadIdx.x == 0) {
        float ssod = scal[0], ssrd = scal[1];
        float a = fmaxf(sqrtf(ssod), EPSF);
        float b = fmaxf(sqrtf(ssrd), EPSF);
        float d2 = ssod / (a * a) - 2.0f * dotr / (a * b) + (float)NN * sv2r / (b * b);
        float lossDistance = sqrtf(fmaxf(d2, 0.f));
        out[0] = lossDistance + scal[4] /*shape*/ + scal[3] /*exceed*/
               + scal[2] /*overlap*/ + scal[5] /*positive*/;
    }
}

extern "C" void kernel_launch(void* const* d_in, const int* in_sizes, int n_in,
                              void* d_out, int out_size, void* d_ws, size_t ws_size,
                              hipStream_t stream) {
    const int*   idIndexes = (const int*)d_in[0];
    const int*   parentIds = (const int*)d_in[1];
    const float* omega     = (const float*)d_in[2];
    const float* childEmb  = (const float*)d_in[3];
    const float* res       = (const float*)d_in[4];
    const float* leaves    = (const float*)d_in[5];
    const float* layerDist = (const float*)d_in[6];
    float* ws  = (float*)d_ws;
    float* out = (float*)d_out;

    hipMemsetAsync(d_ws, 0, (WS_SCAL + 8) * sizeof(float), stream);

    inner_kernel<<<NN / 256, 256, 0, stream>>>(omega, ws + WS_INNER);
    omega_gemm_kernel<<<512, 256, 0, stream>>>(omega, ws + WS_INNER,
                                               ws + WS_CSOD, ws + WS_SCAL);
    pairwise_kernel<<<dim3(NN / 16, NN / 16), 256, 0, stream>>>(
        idIndexes, childEmb, layerDist, ws + WS_CSRD, ws + WS_SCAL);
    elemwise_kernel<<<(NN * SS) / 256, 256, 0, stream>>>(
        idIndexes, parentIds, childEmb, res, leaves, ws + WS_SCAL);
    final_kernel<<<1, 256, 0, stream>>>(ws + WS_CSOD, ws + WS_CSRD,
                                        ws + WS_SCAL, out);
}